// RecurrentGraphNeuralNet_73383811220028
// MI455X (gfx1250) — compile-verified
//
#include <hip/hip_runtime.h>
#include <hip/hip_bf16.h>

typedef float v2f __attribute__((ext_vector_type(2)));
typedef float v8f __attribute__((ext_vector_type(8)));

#define Hc 64
#define Pc 32

// ---------------------------------------------------------------------------
// Kernel 1: zero the accumulator workspace (agg[N*64] + deg[N])
// ---------------------------------------------------------------------------
__global__ void zero_ws_kernel(float* __restrict__ p, long long n) {
    long long i = (long long)blockIdx.x * blockDim.x + threadIdx.x;
    if (i < n) p[i] = 0.0f;
}

// ---------------------------------------------------------------------------
// Kernel 2: edge gather + scatter-add.  16 threads per edge, float4 per thread.
// agg[dst] += x[src]; deg[dst] += 1
// ---------------------------------------------------------------------------
__global__ __launch_bounds__(256)
void edge_scatter_kernel(const float* __restrict__ x,
                         const long long* __restrict__ ei,
                         float* __restrict__ agg,
                         float* __restrict__ deg,
                         int E) {
    long long tid = (long long)blockIdx.x * blockDim.x + threadIdx.x;
    int e = (int)(tid >> 4);
    int g = (int)(tid & 15);
    if (e >= E) return;
    long long src = ei[e];
    long long dst = ei[(long long)E + e];
    const float4 v = *(const float4*)(x + src * Hc + g * 4);
    float* ap = agg + dst * Hc + g * 4;
    unsafeAtomicAdd(ap + 0, v.x);
    unsafeAtomicAdd(ap + 1, v.y);
    unsafeAtomicAdd(ap + 2, v.z);
    unsafeAtomicAdd(ap + 3, v.w);
    if (g == 0) unsafeAtomicAdd(deg + dst, 1.0f);
}

// ---------------------------------------------------------------------------
// Kernel 3: fused  x_new = relu(mean@W + u@B + b) ;  y = x_new@Wp + bp
// Block = 128 threads = 4 waves, 16 nodes per wave (16 | N so tiles are full).
// f32 WMMA 16x16x4:
//   A (16x4): lane<16 -> {A[m][4k],A[m][4k+1]}, lane>=16 -> {A[m][4k+2],A[m][4k+3]}
//   B (4x16): lane<16 -> {B[4k][n],B[4k+1][n]}, lane>=16 -> {B[4k+2][n],B[4k+3][n]}
//   C/D 16x16: vgpr r, lane<16 -> D[r][lane]; lane>=16 -> D[r+8][lane-16]
// ---------------------------------------------------------------------------
__global__ __launch_bounds__(128)
void node_update_kernel(const float* __restrict__ agg,
                        const float* __restrict__ deg,
                        const float* __restrict__ u,
                        const float* __restrict__ W,
                        const float* __restrict__ Bm,
                        const float* __restrict__ bias,
                        const float* __restrict__ Wp,
                        const float* __restrict__ bp,
                        float* __restrict__ xnew,
                        float* __restrict__ y,
                        int N) {
    __shared__ float sW[Hc * Hc];        // 16 KB
    __shared__ float sB[Hc * Hc];        // 16 KB
    __shared__ float sWp[Hc * Pc];       // 8 KB
    __shared__ float sXn[4][16 * Hc];    // 16 KB per-wave x_new staging

    const int t = threadIdx.x;
    for (int i = t; i < Hc * Hc; i += 128) { sW[i] = W[i]; sB[i] = Bm[i]; }
    for (int i = t; i < Hc * Pc; i += 128) sWp[i] = Wp[i];
    __syncthreads();

    const int wave  = t >> 5;
    const int lane  = t & 31;
    const int row   = lane & 15;   // M (for A frags) / N (for B,C frags)
    const int khalf = lane >> 4;   // which K pair within the 4-wide slab
    const int node0 = blockIdx.x * 64 + wave * 16;
    if (node0 >= N) return;        // whole wave exits together (EXEC stays full)

    const int node = node0 + row;
    const float dg = deg[node];
    const float invdeg = 1.0f / (dg > 1.0f ? dg : 1.0f);

    // ---- accumulators for x_new tile (16 x 64 = four 16x16 tiles), bias init
    v8f c[4];
    for (int n = 0; n < 4; ++n) {
        const float bv = bias[n * 16 + row];
        for (int r = 0; r < 8; ++r) c[n][r] = bv;
    }

    // ---- mean @ W
    const float* aggRow = agg + (long long)node * Hc;
    for (int kk = 0; kk < 16; ++kk) {
        const int kc = kk * 4 + khalf * 2;
        v2f a;
        a[0] = aggRow[kc]     * invdeg;
        a[1] = aggRow[kc + 1] * invdeg;
        for (int n = 0; n < 4; ++n) {
            v2f bb;
            bb[0] = sW[kc * Hc + n * 16 + row];
            bb[1] = sW[(kc + 1) * Hc + n * 16 + row];
            c[n] = __builtin_amdgcn_wmma_f32_16x16x4_f32(
                false, a, false, bb, (short)0, c[n], false, false);
        }
    }

    // ---- u @ B
    const float* uRow = u + (long long)node * Hc;
    for (int kk = 0; kk < 16; ++kk) {
        const int kc = kk * 4 + khalf * 2;
        v2f a;
        a[0] = uRow[kc];
        a[1] = uRow[kc + 1];
        for (int n = 0; n < 4; ++n) {
            v2f bb;
            bb[0] = sB[kc * Hc + n * 16 + row];
            bb[1] = sB[(kc + 1) * Hc + n * 16 + row];
            c[n] = __builtin_amdgcn_wmma_f32_16x16x4_f32(
                false, a, false, bb, (short)0, c[n], false, false);
        }
    }

    // ---- ReLU, store x_new, stage tile in LDS for head GEMM
    float* sx = &sXn[wave][0];
    for (int n = 0; n < 4; ++n) {
        for (int r = 0; r < 8; ++r) {
            float v = c[n][r];
            v = v > 0.0f ? v : 0.0f;
            const int M   = r + khalf * 8;
            const int col = n * 16 + row;
            xnew[(long long)(node0 + M) * Hc + col] = v;
            sx[M * Hc + col] = v;
        }
    }
    // per-wave LDS region; DS ops from the same wave are in-order -> no barrier

    // ---- head: y = x_new @ Wp + bp   (16 x 32 = two 16x16 tiles)
    for (int n = 0; n < 2; ++n) {
        v8f cy;
        const float bv = bp[n * 16 + row];
        for (int r = 0; r < 8; ++r) cy[r] = bv;
        for (int kk = 0; kk < 16; ++kk) {
            const int kc = kk * 4 + khalf * 2;
            v2f a;
            a[0] = sx[row * Hc + kc];
            a[1] = sx[row * Hc + kc + 1];
            v2f bb;
            bb[0] = sWp[kc * Pc + n * 16 + row];
            bb[1] = sWp[(kc + 1) * Pc + n * 16 + row];
            cy = __builtin_amdgcn_wmma_f32_16x16x4_f32(
                false, a, false, bb, (short)0, cy, false, false);
        }
        for (int r = 0; r < 8; ++r) {
            const int M = r + khalf * 8;
            y[(long long)(node0 + M) * Pc + n * 16 + row] = cy[r];
        }
    }
}

// ---------------------------------------------------------------------------
extern "C" void kernel_launch(void* const* d_in, const int* in_sizes, int n_in,
                              void* d_out, int out_size, void* d_ws, size_t ws_size,
                              hipStream_t stream) {
    const float*     x  = (const float*)d_in[0];      // [N,64]
    const float*     u  = (const float*)d_in[1];      // [N,64]
    const long long* ei = (const long long*)d_in[2];  // [2,E] int64
    const float*     W  = (const float*)d_in[3];      // [64,64]
    const float*     Bm = (const float*)d_in[4];      // [64,64]
    const float*     b  = (const float*)d_in[5];      // [64]
    const float*     Wp = (const float*)d_in[6];      // [64,32]
    const float*     bp = (const float*)d_in[7];      // [32]

    const int N = in_sizes[0] / Hc;
    const int E = in_sizes[2] / 2;

    float* agg = (float*)d_ws;                 // [N,64]
    float* deg = agg + (long long)N * Hc;      // [N]

    float* xnew = (float*)d_out;               // [N,64]
    float* yout = xnew + (long long)N * Hc;    // [N,32]

    // 1) zero accumulators
    {
        long long n = (long long)N * (Hc + 1);
        int blocks = (int)((n + 255) / 256);
        zero_ws_kernel<<<blocks, 256, 0, stream>>>(agg, n);
    }
    // 2) scatter-add over edges (16 threads / edge)
    {
        long long threads = (long long)E * 16;
        int blocks = (int)((threads + 255) / 256);
        edge_scatter_kernel<<<blocks, 256, 0, stream>>>(x, ei, agg, deg, E);
    }
    // 3) fused WMMA update + head
    {
        int blocks = (N + 63) / 64;
        node_update_kernel<<<blocks, 128, 0, stream>>>(agg, deg, u, W, Bm, b,
                                                       Wp, bp, xnew, yout, N);
    }
}